// CudaSparseMemory_34187939676718
// MI455X (gfx1250) — compile-verified
//
#include <hip/hip_runtime.h>
#include <hip/hip_bf16.h>
#include <math.h>
#include <stdint.h>

#define B_      32
#define INPUT_  512
#define MEM_    32768
#define CELL_   64
#define KTOP    16
#define VIS_    17
#define SPLITS  32
#define ROWS_PER_SPLIT (MEM_ / SPLITS)   /* 1024 */
#define TROWS   128
#define RPAD    68                        /* 64 + 4 pad floats to avoid LDS bank conflicts */
#define NPASS   (ROWS_PER_SPLIT / TROWS)  /* 8 */
#define NCAND   (SPLITS * KTOP)           /* 512 */
#define NEG_INF (-3.402823466e38f)

typedef float v2f __attribute__((ext_vector_type(2)));
typedef float v8f __attribute__((ext_vector_type(8)));

// ---------------------------------------------------------------------------
// Kernel 1: fused linear layers via V_WMMA_F32_16X16X4_F32.
// Y(32 x 160pad) = xi(32x512) @ Wcat^T, per-column segment: [0,64)=tanh->read_query(ws)
// [64,128)=tanh->write_vector, [128,145)=sigmoid->interp_gate, 145=sigmoid->write_gate.
// One wave per 16x16 output tile; grid = 2 (M tiles) x 10 (N tiles).
// ---------------------------------------------------------------------------
__global__ __launch_bounds__(32)
void lin_wmma_kernel(const float* __restrict__ xi,
                     const float* __restrict__ W_rq, const float* __restrict__ b_rq,
                     const float* __restrict__ W_wv, const float* __restrict__ b_wv,
                     const float* __restrict__ W_ig, const float* __restrict__ b_ig,
                     const float* __restrict__ W_wg, const float* __restrict__ b_wg,
                     float* __restrict__ q_ws, float* __restrict__ out_wv,
                     float* __restrict__ out_ig, float* __restrict__ out_wg)
{
    const int lane  = threadIdx.x;
    const int mTile = (int)blockIdx.x & 1;
    const int nTile = (int)blockIdx.x >> 1;
    const int hlf   = lane >> 4;       // 0: K={0,1} / M=r ; 1: K={2,3} / M=r+8
    const int l15   = lane & 15;
    const int arow  = mTile * 16 + l15;   // A-matrix row held by this lane
    const int bcol  = nTile * 16 + l15;   // B-matrix column held by this lane
    const int koff  = hlf * 2;

    // Select weight row (= output column of Y) for the B operand.
    const float* wp;
    if      (bcol < 64)   wp = W_rq + bcol * INPUT_;
    else if (bcol < 128)  wp = W_wv + (bcol - 64) * INPUT_;
    else if (bcol < 145)  wp = W_ig + (bcol - 128) * INPUT_;
    else if (bcol == 145) wp = W_wg;
    else                  wp = W_rq;   // pad columns: harmless valid memory, result discarded

    const float* ap = xi + arow * INPUT_ + koff;
    const float* bp = wp + koff;

    v8f acc = {0.f, 0.f, 0.f, 0.f, 0.f, 0.f, 0.f, 0.f};
    for (int k = 0; k < INPUT_; k += 4) {
        v2f a; a.x = ap[k]; a.y = ap[k + 1];
        v2f b; b.x = bp[k]; b.y = bp[k + 1];
        acc = __builtin_amdgcn_wmma_f32_16x16x4_f32(
            /*neg_a=*/false, a, /*neg_b=*/false, b,
            /*c_mod=*/(short)0, acc, /*reuse_a=*/false, /*reuse_b=*/false);
    }

    const int dcol = bcol;   // D-matrix: lane holds column (lane&15), rows r (+8 if hlf)
    if (dcol < 146) {
        float bias;
        if      (dcol < 64)  bias = b_rq[dcol];
        else if (dcol < 128) bias = b_wv[dcol - 64];
        else if (dcol < 145) bias = b_ig[dcol - 128];
        else                 bias = b_wg[0];
        #pragma unroll
        for (int r = 0; r < 8; ++r) {
            int drow = mTile * 16 + r + hlf * 8;   // 0..31
            float y = acc[r] + bias;
            if (dcol < 128) {
                y = tanhf(y);
                if (dcol < 64) q_ws[drow * CELL_ + dcol]          = y;
                else           out_wv[drow * CELL_ + (dcol - 64)] = y;
            } else {
                y = 1.0f / (1.0f + expf(-y));
                if (dcol < 145) out_ig[drow * VIS_ + (dcol - 128)] = y;
                else            out_wg[drow]                       = y;
            }
        }
    }
}

// ---------------------------------------------------------------------------
// Kernel 2: stream memory rows, compute sims = <q, row>, keep per-split top-16.
// Grid (SPLITS, B); 128 threads. Tiles staged into LDS with CDNA5 async
// global->LDS copies (ASYNCcnt), non-temporal since the 268MB stream is
// touched exactly once (larger than the 192MB L2).
// ---------------------------------------------------------------------------
__global__ __launch_bounds__(128)
void sims_topk_kernel(const float* __restrict__ memory,
                      const float* __restrict__ q_ws,
                      float* __restrict__ candV, int* __restrict__ candI)
{
    __shared__ float tile[TROWS * RPAD];        // ~34.8 KB
    __shared__ float qs[CELL_];
    __shared__ float scores[ROWS_PER_SPLIT];    // 4 KB
    __shared__ float rv[128];
    __shared__ int   ri[128];

    const int tid   = threadIdx.x;
    const int split = blockIdx.x;
    const int batch = blockIdx.y;

    if (tid < CELL_) qs[tid] = q_ws[batch * CELL_ + tid];

    const float* memB = memory + (size_t)batch * MEM_ * CELL_
                               + (size_t)split * ROWS_PER_SPLIT * CELL_;
    __syncthreads();

    for (int pass = 0; pass < NPASS; ++pass) {
        const float* src = memB + pass * TROWS * CELL_;
        // Async copy of the 128x64 f32 tile into LDS (16B per lane per issue).
        #pragma unroll
        for (int i = 0; i < (TROWS * CELL_ / 4) / 128; ++i) {  // 16 issues per thread
            int f   = tid + i * 128;           // float4 index within tile
            int row = f >> 4;
            int c4  = f & 15;
            uint32_t lds = (uint32_t)(uintptr_t)(&tile[row * RPAD + c4 * 4]);
            uint64_t ga  = (uint64_t)(uintptr_t)(src + f * 4);
            asm volatile("global_load_async_to_lds_b128 %0, %1, off th:TH_LOAD_NT"
                         :: "v"(lds), "v"(ga) : "memory");
        }
        asm volatile("s_wait_asynccnt 0x0" ::: "memory");
        __syncthreads();

        float acc = 0.f;
        const float* trow = &tile[tid * RPAD];
        #pragma unroll
        for (int c4 = 0; c4 < 16; ++c4) {
            float4 m  = *(const float4*)&trow[c4 * 4];
            float4 q4 = *(const float4*)&qs[c4 * 4];
            acc += m.x * q4.x + m.y * q4.y + m.z * q4.z + m.w * q4.w;
        }
        scores[pass * TROWS + tid] = acc;
        __syncthreads();
    }

    // iterative argmax x16 (descending; ties -> smaller index, matching lax.top_k)
    for (int k = 0; k < KTOP; ++k) {
        float best = NEG_INF; int bi = 0x7fffffff;
        #pragma unroll
        for (int i = 0; i < NPASS; ++i) {
            int idx = tid + i * 128;
            float s = scores[idx];
            if (s > best || (s == best && idx < bi)) { best = s; bi = idx; }
        }
        rv[tid] = best; ri[tid] = bi;
        __syncthreads();
        for (int off = 64; off > 0; off >>= 1) {
            if (tid < off) {
                float ov = rv[tid + off]; int oi = ri[tid + off];
                if (ov > rv[tid] || (ov == rv[tid] && oi < ri[tid])) { rv[tid] = ov; ri[tid] = oi; }
            }
            __syncthreads();
        }
        if (tid == 0) {
            int slot = (batch * SPLITS + split) * KTOP + k;
            candV[slot] = rv[0];
            candI[slot] = split * ROWS_PER_SPLIT + ri[0];
            scores[ri[0]] = NEG_INF;
        }
        __syncthreads();
    }
}

// ---------------------------------------------------------------------------
// Kernel 3: zero the read_weights output (4 MB) before the scatter.
// ---------------------------------------------------------------------------
__global__ void zero_kernel(float4* __restrict__ p, int n4)
{
    int i = blockIdx.x * blockDim.x + threadIdx.x;
    if (i < n4) p[i] = make_float4(0.f, 0.f, 0.f, 0.f);
}

// ---------------------------------------------------------------------------
// Kernel 4: per batch — merge 512 candidates to global top-16 (descending),
// append least-used slot, gather visible memory, softmax attention,
// read_vectors, scatter read_weights. One block per batch.
// ---------------------------------------------------------------------------
__global__ __launch_bounds__(128)
void finalize_kernel(const float* __restrict__ memory,
                     const int* __restrict__ least_used,
                     const float* __restrict__ q_ws,
                     const float* __restrict__ candV, const int* __restrict__ candI,
                     float* __restrict__ out_rv, float* __restrict__ out_rw,
                     float* __restrict__ out_vis)
{
    __shared__ float cv[NCAND];
    __shared__ int   ci[NCAND];
    __shared__ float rv[128];
    __shared__ int   ri[128];
    __shared__ int   rs[128];
    __shared__ float qs[CELL_];
    __shared__ int   pos[VIS_];
    __shared__ float vm[VIS_ * 68];   // padded rows
    __shared__ float att[VIS_];

    const int tid   = threadIdx.x;
    const int batch = blockIdx.x;

    for (int i = tid; i < NCAND; i += 128) {
        cv[i] = candV[batch * NCAND + i];
        ci[i] = candI[batch * NCAND + i];
    }
    if (tid < CELL_) qs[tid] = q_ws[batch * CELL_ + tid];
    __syncthreads();

    for (int k = 0; k < KTOP; ++k) {
        float best = NEG_INF; int bi = 0x7fffffff; int bslot = 0;
        #pragma unroll
        for (int i = 0; i < NCAND / 128; ++i) {
            int slot = tid + i * 128;
            float s = cv[slot]; int mi = ci[slot];
            if (s > best || (s == best && mi < bi)) { best = s; bi = mi; bslot = slot; }
        }
        rv[tid] = best; ri[tid] = bi; rs[tid] = bslot;
        __syncthreads();
        for (int off = 64; off > 0; off >>= 1) {
            if (tid < off) {
                float ov = rv[tid + off];
                if (ov > rv[tid] || (ov == rv[tid] && ri[tid + off] < ri[tid])) {
                    rv[tid] = ov; ri[tid] = ri[tid + off]; rs[tid] = rs[tid + off];
                }
            }
            __syncthreads();
        }
        if (tid == 0) { pos[k] = ri[0]; cv[rs[0]] = NEG_INF; }
        __syncthreads();
    }
    if (tid == 0) pos[KTOP] = least_used[batch];
    __syncthreads();

    // gather visible memory: to output and LDS
    for (int f = tid; f < VIS_ * CELL_; f += 128) {
        int v = f >> 6, c = f & 63;
        float x = memory[(size_t)batch * MEM_ * CELL_ + (size_t)pos[v] * CELL_ + c];
        vm[v * 68 + c] = x;
        out_vis[batch * (VIS_ * CELL_) + f] = x;
    }
    __syncthreads();

    // attention scores over the 17 visible cells
    if (tid < VIS_) {
        float s = 0.f;
        #pragma unroll
        for (int c = 0; c < CELL_; ++c) s += qs[c] * vm[tid * 68 + c];
        att[tid] = s;
    }
    __syncthreads();
    if (tid == 0) {
        float mx = att[0];
        for (int v = 1; v < VIS_; ++v) mx = fmaxf(mx, att[v]);
        float sum = 0.f;
        for (int v = 0; v < VIS_; ++v) { float e = expf(att[v] - mx); att[v] = e; sum += e; }
        float inv = 1.f / sum;
        for (int v = 0; v < VIS_; ++v) att[v] *= inv;
    }
    __syncthreads();

    // read_vectors = att @ visible_memory
    if (tid < CELL_) {
        float r = 0.f;
        #pragma unroll
        for (int v = 0; v < VIS_; ++v) r += att[v] * vm[v * 68 + tid];
        out_rv[batch * CELL_ + tid] = r;
    }
    // scatter read weights; least-used written last so it wins on duplicate position
    if (tid < KTOP) out_rw[(size_t)batch * MEM_ + pos[tid]] = att[tid];
    __syncthreads();
    if (tid == 0) out_rw[(size_t)batch * MEM_ + pos[KTOP]] = att[KTOP];
}

// ---------------------------------------------------------------------------
extern "C" void kernel_launch(void* const* d_in, const int* in_sizes, int n_in,
                              void* d_out, int out_size, void* d_ws, size_t ws_size,
                              hipStream_t stream)
{
    const float* xi     = (const float*)d_in[0];
    const float* memory = (const float*)d_in[1];
    const int*   lum    = (const int*)d_in[2];
    const float* W_rq   = (const float*)d_in[3];
    const float* b_rq   = (const float*)d_in[4];
    const float* W_wv   = (const float*)d_in[5];
    const float* b_wv   = (const float*)d_in[6];
    const float* W_ig   = (const float*)d_in[7];
    const float* b_ig   = (const float*)d_in[8];
    const float* W_wg   = (const float*)d_in[9];
    const float* b_wg   = (const float*)d_in[10];

    float* out     = (float*)d_out;
    float* out_rv  = out;                       // [32,1,64]
    float* out_rw  = out + 2048;                // [32,32768]
    float* out_vis = out + 2048 + 1048576;      // [32,17,64]
    float* out_wv  = out_vis + 34816;           // [32,1,64]
    float* out_ig  = out_wv + 2048;             // [32,17]
    float* out_wg  = out_ig + 544;              // [32,1]

    float* ws    = (float*)d_ws;
    float* q_ws  = ws;                          // 32*64 read_query
    float* candV = ws + 2048;                   // 32*512 candidate values
    int*   candI = (int*)(ws + 2048 + NCAND * B_);  // 32*512 candidate indices

    lin_wmma_kernel<<<20, 32, 0, stream>>>(xi, W_rq, b_rq, W_wv, b_wv,
                                           W_ig, b_ig, W_wg, b_wg,
                                           q_ws, out_wv, out_ig, out_wg);

    dim3 grid(SPLITS, B_);
    sims_topk_kernel<<<grid, 128, 0, stream>>>(memory, q_ws, candV, candI);

    zero_kernel<<<(262144 + 255) / 256, 256, 0, stream>>>((float4*)out_rw, 262144);

    finalize_kernel<<<B_, 128, 0, stream>>>(memory, lum, q_ws, candV, candI,
                                            out_rv, out_rw, out_vis);
}